// RBFNSemanticLayer_17274358464773
// MI455X (gfx1250) — compile-verified
//
#include <hip/hip_runtime.h>

typedef float v2f __attribute__((ext_vector_type(2)));
typedef float v8f __attribute__((ext_vector_type(8)));

#define NPTS   (64 * 2048)   // B * P
#define KN     16            // neighbors
#define NC     16            // RBF centers
#define SFD    64            // spatial feature dim == OUT_F

// ---------------------------------------------------------------------------
// Kernel 1: fused RBF + K-sum + spatial K-sum + elementwise multiply.
// One wave32 per point pt.  Writes x[pt][0..63] (fp32) into workspace.
//   s[c]  = sum_k exp(-||feat[pt,k] - center[c]||^2 / 0.08)      (16 vals)
//   sp[f] = sum_k spatial[pt,k,f]                                 (64 vals)
//   x[f]  = (sum_c s[c] * rbfn_weight[c,f]) * sp[f]
// ---------------------------------------------------------------------------
__global__ __launch_bounds__(256) void rbfn_fuse_kernel(
    const float* __restrict__ features,     // (NPTS, K, 3)
    const float* __restrict__ spatial,      // (NPTS, K, 64)
    const float* __restrict__ centers,      // (3, 16)
    const float* __restrict__ rbfn_weight,  // (16, 64)
    float* __restrict__ xout)               // (NPTS, 64)
{
    const int lane = threadIdx.x & 31;
    const int pt   = blockIdx.x * 8 + (threadIdx.x >> 5);
    const int c    = lane & 15;  // both half-waves duplicate the RBF work

    // this lane's RBF center (centers stored as (dim, n_centers))
    const float cx = centers[0 * NC + c];
    const float cy = centers[1 * NC + c];
    const float cz = centers[2 * NC + c];

    // s[c] = sum_k exp2( dn2 * (-12.5 * log2 e) )
    const float* fp = features + (size_t)pt * (KN * 3);
    const float kexp = -18.033688011112043f;  // -(1/0.08) * log2(e)
    float s = 0.0f;
    #pragma unroll
    for (int k = 0; k < KN; ++k) {
        float dx = fp[k * 3 + 0] - cx;
        float dy = fp[k * 3 + 1] - cy;
        float dz = fp[k * 3 + 2] - cz;
        float dn2 = dx * dx + dy * dy + dz * dz;
        s += __builtin_amdgcn_exp2f(dn2 * kexp);
    }

    // spatial K-sum: lane owns features {2*lane, 2*lane+1}; stream non-temporal
    const float* spp = spatial + (size_t)pt * (KN * SFD) + 2 * lane;
    v2f sp = {0.0f, 0.0f};
    #pragma unroll
    for (int k = 0; k < KN; ++k) {
        v2f v = __builtin_nontemporal_load((const v2f*)(spp + k * SFD));
        sp += v;
    }

    // fl[f] = sum_c s[c] * rbfn_weight[c][f]   (broadcast s via wave shuffle)
    v2f fl = {0.0f, 0.0f};
    #pragma unroll
    for (int cc = 0; cc < NC; ++cc) {
        float sc = __shfl(s, cc, 32);
        v2f rw = *(const v2f*)(rbfn_weight + cc * SFD + 2 * lane);
        fl += sc * rw;
    }

    v2f x = fl * sp;
    *(v2f*)(xout + (size_t)pt * SFD + 2 * lane) = x;
}

// ---------------------------------------------------------------------------
// Kernel 2: per-batch GEMM  out[b] = x[b] (2048x64) @ w[b] (64x64)  in fp32
// via V_WMMA_F32_16X16X4_F32.  One wave per 16-row block: 4 N-tiles of 16,
// K-loop of 16 steps (4 K-values per WMMA).
// A layout (16x4 f32): lanes 0-15 hold K={4kk,4kk+1}, lanes 16-31 K={4kk+2,4kk+3}
// B layout (4x16 f32): VGPR0 = row K (this half), VGPR1 = row K+1, col = lane&15
// C/D layout (16x16 f32): VGPR r -> row (khalf*8 + r), col = lane&15
// ---------------------------------------------------------------------------
__global__ __launch_bounds__(256) void batched_gemm_wmma(
    const float* __restrict__ x,   // (NPTS, 64)
    const float* __restrict__ w,   // (B, 64, 64)
    float* __restrict__ out)       // (NPTS, 64)
{
    const int lane     = threadIdx.x & 31;
    const int rowBlock = blockIdx.x * 8 + (threadIdx.x >> 5); // of 8192
    const int batch    = rowBlock >> 7;     // 128 row-blocks per batch (P/16)
    const int m        = lane & 15;
    const int khalf    = lane >> 4;

    const float* xrow = x + (size_t)(rowBlock * 16 + m) * 64;
    const float* wb   = w + (size_t)batch * 64 * 64;

    v8f acc0 = {}, acc1 = {}, acc2 = {}, acc3 = {};

    #pragma unroll
    for (int kk = 0; kk < 16; ++kk) {
        const int k0 = kk * 4 + khalf * 2;
        v2f a = *(const v2f*)(xrow + k0);             // {A[m][k0], A[m][k0+1]}
        const float* w0 = wb + k0 * 64 + m;
        v2f b0 = { w0[0],  w0[64 + 0]  };             // N-tile 0, rows k0,k0+1
        v2f b1 = { w0[16], w0[64 + 16] };             // N-tile 1
        v2f b2 = { w0[32], w0[64 + 32] };             // N-tile 2
        v2f b3 = { w0[48], w0[64 + 48] };             // N-tile 3
        acc0 = __builtin_amdgcn_wmma_f32_16x16x4_f32(false, a, false, b0, (short)0, acc0, false, false);
        acc1 = __builtin_amdgcn_wmma_f32_16x16x4_f32(false, a, false, b1, (short)0, acc1, false, false);
        acc2 = __builtin_amdgcn_wmma_f32_16x16x4_f32(false, a, false, b2, (short)0, acc2, false, false);
        acc3 = __builtin_amdgcn_wmma_f32_16x16x4_f32(false, a, false, b3, (short)0, acc3, false, false);
    }

    // scatter D: VGPR r -> global row rowBlock*16 + khalf*8 + r, col nt*16 + m
    float* orow = out + (size_t)(rowBlock * 16 + khalf * 8) * 64 + m;
    #pragma unroll
    for (int r = 0; r < 8; ++r) {
        float* o = orow + (size_t)r * 64;
        __builtin_nontemporal_store(acc0[r], o + 0);
        __builtin_nontemporal_store(acc1[r], o + 16);
        __builtin_nontemporal_store(acc2[r], o + 32);
        __builtin_nontemporal_store(acc3[r], o + 48);
    }
}

// ---------------------------------------------------------------------------
extern "C" void kernel_launch(void* const* d_in, const int* in_sizes, int n_in,
                              void* d_out, int out_size, void* d_ws, size_t ws_size,
                              hipStream_t stream) {
    const float* features = (const float*)d_in[0];  // (B,P,K,3)
    const float* spatial  = (const float*)d_in[1];  // (B,P,K,64)
    const float* centers  = (const float*)d_in[2];  // (3,16)
    const float* rbfnw    = (const float*)d_in[3];  // (16,64)
    const float* w        = (const float*)d_in[4];  // (64,64,64)
    // d_in[5] is K (==16), baked into the kernels.

    float* xws = (float*)d_ws;                      // NPTS*64 fp32 = 33.5 MB

    // Kernel 1: one wave per point, 8 waves per 256-thread block.
    rbfn_fuse_kernel<<<NPTS / 8, 256, 0, stream>>>(features, spatial, centers,
                                                   rbfnw, xws);
    // Kernel 2: one wave per 16-row block, 8 waves per block.
    batched_gemm_wmma<<<(NPTS / 16) / 8, 256, 0, stream>>>(xws, w, (float*)d_out);
}